// VolumeRenderer_36902359007484
// MI455X (gfx1250) — compile-verified
//
// NeRF-style fused volume renderer for MI455X (gfx1250).
// Compute: 5.7 GFLOP; HBM: ~25 MB (~1us @ 23.3 TB/s); emb table (12.8 MB) is
// L2-resident, 1M-row gather served from L2 -> fuse MLP + compositing in one
// kernel and use full-precision V_WMMA_F32_16X16X4_F32 (matrix pipe is not the
// bottleneck, so no need to trade precision for fp16/fp8 rate).
#include <hip/hip_runtime.h>

typedef __attribute__((ext_vector_type(2))) float v2f;
typedef __attribute__((ext_vector_type(4))) float v4f;
typedef __attribute__((ext_vector_type(8))) float v8f;

#if __has_builtin(__builtin_amdgcn_wmma_f32_16x16x4_f32)
#define USE_WMMA_F32 1
#else
#define USE_WMMA_F32 0
#endif

namespace {
constexpr int kS = 64;         // samples per ray
constexpr int kD = 32;         // embedding dim
constexpr int kH = 64;         // hidden dim
constexpr int kK = 38;         // D + 6 (feat | pos | dir)
constexpr int kKPad = 40;      // padded K, multiple of 4
constexpr int kW1Stride = 72;  // LDS row stride in words (bank-friendly)
constexpr int kWaves = 8;      // waves per block (wave32)
}

// One wave per ray. Computes Hᵀ = W1ᵀ·Xᵀ with 16x16x4 f32 WMMA:
//   C tile: M = hidden (16t..16t+15), N = 16 samples of the ray.
//   A = W1ᵀ slice from LDS, B = Xᵀ slice (emb gather + pos/dir).
// Then per-lane 64->4 GEMM (each lane owns one sample's hidden vector half),
// then a shfl-based exclusive scan for the transmittance compositing.
__global__ __launch_bounds__(256) void nerf_fused_kernel(
    const float* __restrict__ ray_start, const float* __restrict__ ray_dir,
    const float* __restrict__ sampled_depth, const float* __restrict__ sampled_dists,
    const float* __restrict__ emb, const float* __restrict__ w1,
    const float* __restrict__ b1, const float* __restrict__ w2,
    const float* __restrict__ b2, const int* __restrict__ sampled_idx,
    float* __restrict__ out_rgb, float* __restrict__ out_depth,
    float* __restrict__ out_missed, float* __restrict__ ws_var, int N) {
  __shared__ __align__(16) float lw1[kKPad * kW1Stride];
  __shared__ __align__(16) float lb1[kH];
  __shared__ __align__(16) float lw2[kH * 4];
  __shared__ float svar[kWaves];

  const int tid = threadIdx.x;
  // Stage W1 (38x64, zero-padded to 40 rows), b1, w2 into LDS.
  for (int i = tid; i < kKPad * kH; i += 256) {
    const int r = i >> 6, c = i & 63;
    lw1[r * kW1Stride + c] = (r < kK) ? w1[r * kH + c] : 0.0f;
  }
  for (int i = tid; i < kH; i += 256) lb1[i] = b1[i];
  for (int i = tid; i < kH * 4; i += 256) lw2[i] = w2[i];
  __syncthreads();

  const int wave = tid >> 5;
  const int lane = tid & 31;
  const int li = lane & 15;  // sample / column index within tile
  const int hl = lane >> 4;  // lane-half (selects K pair / hidden half)
  const int ray = blockIdx.x * kWaves + wave;

  float acc_d = 0.f, acc_d2 = 0.f;

  if (ray < N) {
    // Ray constants (wave-uniform).
    const float rs0 = ray_start[ray * 3 + 0], rs1 = ray_start[ray * 3 + 1],
                rs2 = ray_start[ray * 3 + 2];
    const float rd0 = ray_dir[ray * 3 + 0], rd1 = ray_dir[ray * 3 + 1],
                rd2 = ray_dir[ray * 3 + 2];
    const float b20 = b2[0], b21 = b2[1], b22 = b2[2], b23 = b2[3];

    float carry = 0.f;  // exclusive free-energy prefix carried across tiles
    float acc_p = 0.f, acc_r0 = 0.f, acc_r1 = 0.f, acc_r2 = 0.f;

    for (int ts = 0; ts < kS / 16; ++ts) {
      const int g = ray * kS + ts * 16 + li;
      const float dep = sampled_depth[g];
      const float dst = sampled_dists[g];
      const int idx = sampled_idx[g];
      const bool valid = idx >= 0;
      const float* erow = emb + (size_t)(valid ? idx : 0) * kD;

      const float p0 = rs0 + rd0 * dep;
      const float p1 = rs1 + rd1 * dep;
      const float p2 = rs2 + rd2 * dep;

      v8f acc[4] = {};  // Hᵀ tiles: hidden 16t..16t+15  x  16 samples
#if USE_WMMA_F32
      // B operand (Xᵀ): lane holds x[sample=li, k = 4*kk + 2*hl + {0,1}].
      v2f Bv[10];
#pragma unroll
      for (int kk = 0; kk < 8; ++kk) {
        Bv[kk] = *(const v2f*)(erow + kk * 4 + hl * 2);  // b64 L2 gather
      }
      Bv[8].x = hl ? p2 : p0;   // k=34 : k=32
      Bv[8].y = hl ? rd0 : p1;  // k=35 : k=33
      Bv[9].x = hl ? 0.f : rd1; // pad  : k=36
      Bv[9].y = hl ? 0.f : rd2; // pad  : k=37
#pragma unroll
      for (int kk = 0; kk < 10; ++kk) {
        const int rowA = kk * 4 + hl * 2;
#pragma unroll
        for (int t = 0; t < 4; ++t) {
          // A operand (W1ᵀ): lane holds w1[k, hid = 16t + li], k = rowA + {0,1}.
          v2f a;
          a.x = lw1[rowA * kW1Stride + t * 16 + li];
          a.y = lw1[(rowA + 1) * kW1Stride + t * 16 + li];
          acc[t] = __builtin_amdgcn_wmma_f32_16x16x4_f32(
              false, a, false, Bv[kk], (short)0, acc[t], false, false);
        }
      }
#else
      // Scalar fallback (diagnostic only): same acc[] layout via per-lane dots.
      float xv[kK];
#pragma unroll
      for (int k = 0; k < kD; ++k) xv[k] = erow[k];
      xv[32] = p0; xv[33] = p1; xv[34] = p2;
      xv[35] = rd0; xv[36] = rd1; xv[37] = rd2;
#pragma unroll
      for (int t = 0; t < 4; ++t) {
#pragma unroll
        for (int v = 0; v < 8; ++v) {
          const int hidx = t * 16 + hl * 8 + v;
          float sacc = 0.f;
          for (int k = 0; k < kK; ++k) sacc += xv[k] * lw1[k * kW1Stride + hidx];
          acc[t][v] = sacc;
        }
      }
#endif
      // bias + ReLU, then 64->4 GEMM. C layout: lane holds rows
      // hid = 16t + 8*hl + v for its sample (col li) -> per-lane FMAs + one
      // xor-16 shuffle combine the two hidden halves.
      float o0 = 0.f, o1 = 0.f, o2 = 0.f, o3 = 0.f;
#pragma unroll
      for (int t = 0; t < 4; ++t) {
#pragma unroll
        for (int v = 0; v < 8; ++v) {
          const int hidx = t * 16 + hl * 8 + v;
          float hv = acc[t][v] + lb1[hidx];
          hv = hv > 0.f ? hv : 0.f;
          const v4f w = *(const v4f*)&lw2[hidx * 4];  // LDS broadcast
          o0 += hv * w.x; o1 += hv * w.y; o2 += hv * w.z; o3 += hv * w.w;
        }
      }
      o0 += __shfl_xor(o0, 16, 32);
      o1 += __shfl_xor(o1, 16, 32);
      o2 += __shfl_xor(o2, 16, 32);
      o3 += __shfl_xor(o3, 16, 32);

      const float sigma = o0 + b20;
      const float fe = valid ? (sigma > 0.f ? sigma : 0.f) * dst : 0.f;
      const float t0 = valid ? o1 + b21 : 0.f;
      const float t1 = valid ? o2 + b22 : 0.f;
      const float t2 = valid ? o3 + b23 : 0.f;

      // Inclusive scan of fe over the 16 samples of this tile (Hillis-Steele).
      float incl = fe;
#pragma unroll
      for (int d = 1; d < 16; d <<= 1) {
        const float up = __shfl_up(incl, d, 16);
        if (li >= d) incl += up;
      }
      const float cum = carry + (incl - fe);     // exclusive prefix + carry
      const float alpha = 1.f - __expf(-fe);
      const float trans = __expf(-cum);
      const float p = alpha * trans;
      acc_p += p;
      acc_d += dep * p;
      acc_d2 += dep * dep * p;
      acc_r0 += t0 * p; acc_r1 += t1 * p; acc_r2 += t2 * p;
      carry += __shfl(incl, 15, 16);             // tile total -> next tile
    }

    // Reduce the 16 per-sample accumulators (both halves identical).
#pragma unroll
    for (int d = 1; d < 16; d <<= 1) {
      acc_p += __shfl_xor(acc_p, d, 32);
      acc_d += __shfl_xor(acc_d, d, 32);
      acc_d2 += __shfl_xor(acc_d2, d, 32);
      acc_r0 += __shfl_xor(acc_r0, d, 32);
      acc_r1 += __shfl_xor(acc_r1, d, 32);
      acc_r2 += __shfl_xor(acc_r2, d, 32);
    }
    if (lane == 0) {
      out_rgb[ray * 3 + 0] = acc_r0;
      out_rgb[ray * 3 + 1] = acc_r1;
      out_rgb[ray * 3 + 2] = acc_r2;
      out_depth[ray] = acc_d;
      out_missed[ray] = 1.f - acc_p;
    }
  }

  // Deterministic var_loss: per-block partial sums (no float atomics).
  if (lane == 0) svar[wave] = (ray < N) ? (acc_d2 - acc_d * acc_d) : 0.f;
  __syncthreads();
  if (tid == 0) {
    float sum = 0.f;
    for (int w = 0; w < kWaves; ++w) sum += svar[w];
    ws_var[blockIdx.x] = sum;
  }
}

__global__ __launch_bounds__(256) void var_reduce_kernel(
    const float* __restrict__ ws, float* __restrict__ out_var, int nblocks,
    float invN) {
  __shared__ float sm[256];
  const int t = threadIdx.x;
  float a = 0.f;
  for (int i = t; i < nblocks; i += 256) a += ws[i];  // fixed order
  sm[t] = a;
  __syncthreads();
  for (int s = 128; s > 0; s >>= 1) {
    if (t < s) sm[t] += sm[t + s];
    __syncthreads();
  }
  if (t == 0) out_var[0] = sm[0] * invN;
}

extern "C" void kernel_launch(void* const* d_in, const int* in_sizes, int n_in,
                              void* d_out, int out_size, void* d_ws,
                              size_t ws_size, hipStream_t stream) {
  (void)n_in; (void)out_size; (void)ws_size;
  const float* ray_start = (const float*)d_in[0];
  const float* ray_dir = (const float*)d_in[1];
  const float* sampled_depth = (const float*)d_in[2];
  const float* sampled_dists = (const float*)d_in[3];
  const float* emb = (const float*)d_in[4];
  const float* w1 = (const float*)d_in[5];
  const float* b1 = (const float*)d_in[6];
  const float* w2 = (const float*)d_in[7];
  const float* b2 = (const float*)d_in[8];
  const int* sampled_idx = (const int*)d_in[9];

  const int N = in_sizes[0] / 3;  // 16384 rays
  float* out = (float*)d_out;     // [rgb(N*3) | depth(N) | missed(N) | var(1)]
  float* out_rgb = out;
  float* out_depth = out + (size_t)N * 3;
  float* out_missed = out + (size_t)N * 4;
  float* out_var = out + (size_t)N * 5;
  float* ws = (float*)d_ws;       // nblocks floats (8 KB) of var partials

  const int nblocks = (N + kWaves - 1) / kWaves;
  nerf_fused_kernel<<<nblocks, 256, 0, stream>>>(
      ray_start, ray_dir, sampled_depth, sampled_dists, emb, w1, b1, w2, b2,
      sampled_idx, out_rgb, out_depth, out_missed, ws, N);
  var_reduce_kernel<<<1, 256, 0, stream>>>(ws, out_var, nblocks,
                                           1.0f / (float)N);
}